// MoonDiscriminator_592705487395
// MI455X (gfx1250) — compile-verified
//
#include <hip/hip_runtime.h>
#include <hip/hip_bf16.h>
#include <math.h>

// ---------------- problem constants ----------------
#define GG    4096          // graphs
#define NPG   198           // nodes per graph
#define EPG   1584          // edges per graph
#define NTOT  (GG * NPG)    // 811008
#define ETOT  (GG * EPG)    // 6488064

typedef __attribute__((ext_vector_type(2))) float v2f;
typedef __attribute__((ext_vector_type(8))) float v8f;

// =====================================================================
// Fused 5-layer GraphConv: one workgroup per graph, everything in LDS.
// h = relu( lin_rel(segment_sum(x[src]*ew, dst)) + lin_root(x) )
// =====================================================================
template<int CI, int CO>
__device__ __forceinline__ void conv_layer(
    const float* __restrict__ g_wr, const float* __restrict__ g_br,
    const float* __restrict__ g_wroot,
    const float* bin, float* bout, float* bagg,
    const int* s_edge, const float* s_ew,
    float* s_wr, float* s_wroot, float* s_br, int tid)
{
    // stage tiny weights in LDS
    for (int i = tid; i < CO * CI; i += 256) {
        s_wr[i]    = g_wr[i];
        s_wroot[i] = g_wroot[i];
    }
    if (tid < CO) s_br[tid] = g_br[tid];
    // zero aggregation buffer
    for (int i = tid; i < NPG * CI; i += 256) bagg[i] = 0.f;
    __syncthreads();

    // scatter-add messages via LDS atomics (ds_add_f32)
    for (int e = tid; e < EPG; e += 256) {
        int   ed = s_edge[e];
        int   sb = (ed >> 16)    * CI;
        int   db = (ed & 0xffff) * CI;
        float w  = s_ew[e];
        #pragma unroll
        for (int c = 0; c < CI; ++c)
            atomicAdd(&bagg[db + c], bin[sb + c] * w);
    }
    __syncthreads();

    // dense transform + relu
    for (int idx = tid; idx < NPG * CO; idx += 256) {
        int node = idx / CO;
        int o    = idx - node * CO;
        float acc = s_br[o];
        const float* wr  = s_wr    + o * CI;
        const float* wro = s_wroot + o * CI;
        const float* xa  = bagg + node * CI;
        const float* xi  = bin  + node * CI;
        #pragma unroll
        for (int c = 0; c < CI; ++c)
            acc += xa[c] * wr[c] + xi[c] * wro[c];
        bout[idx] = fmaxf(acc, 0.f);
    }
    __syncthreads();
}

__global__ __launch_bounds__(256) void conv_fused_kernel(
    const float* __restrict__ x, const int* __restrict__ ei,
    const float* __restrict__ ew,
    const float* wr0, const float* br0, const float* wo0,
    const float* wr1, const float* br1, const float* wo1,
    const float* wr2, const float* br2, const float* wo2,
    const float* wr3, const float* br3, const float* wo3,
    const float* wr4, const float* br4, const float* wo4,
    float* __restrict__ H0)
{
    __shared__ int   s_edge[EPG];        //  6336 B (packed src<<16|dst)
    __shared__ float s_ew[EPG];          //  6336 B
    __shared__ float bufA[NPG * 15];     // 11880 B
    __shared__ float bufB[NPG * 20];     // 15840 B
    __shared__ float bufG[NPG * 20];     // 15840 B
    __shared__ float s_wr[20 * 20];
    __shared__ float s_wroot[20 * 20];
    __shared__ float s_br[32];           // total ~58.3 KB -> 5 WGs/WGP

    const int g   = blockIdx.x;
    const int tid = threadIdx.x;

    // load this graph's node features (198 x 2)
    for (int i = tid; i < NPG * 2; i += 256)
        bufA[i] = x[(size_t)g * NPG * 2 + i];
    // load this graph's edges once (reused by all 5 layers)
    for (int e = tid; e < EPG; e += 256) {
        int eg = g * EPG + e;
        int s  = ei[eg]        - g * NPG;   // local src
        int d  = ei[ETOT + eg] - g * NPG;   // local dst
        s_edge[e] = (s << 16) | d;
        s_ew[e]   = ew[eg];
    }
    __syncthreads();

    conv_layer< 2, 20>(wr0, br0, wo0, bufA, bufB, bufG, s_edge, s_ew, s_wr, s_wroot, s_br, tid);
    conv_layer<20, 15>(wr1, br1, wo1, bufB, bufA, bufG, s_edge, s_ew, s_wr, s_wroot, s_br, tid);
    conv_layer<15, 10>(wr2, br2, wo2, bufA, bufB, bufG, s_edge, s_ew, s_wr, s_wroot, s_br, tid);
    conv_layer<10,  5>(wr3, br3, wo3, bufB, bufA, bufG, s_edge, s_ew, s_wr, s_wroot, s_br, tid);
    conv_layer< 5,  2>(wr4, br4, wo4, bufA, bufB, bufG, s_edge, s_ew, s_wr, s_wroot, s_br, tid);

    // graph g's MLP input row = its 198 nodes x 2 channels, flat
    for (int i = tid; i < NPG * 2; i += 256)
        H0[(size_t)g * NPG * 2 + i] = bufB[i];
}

// =====================================================================
// MLP layer as f32 WMMA GEMM: C[M,N] = relu(A[M,K] @ W[N,K]^T + b)
// One wave per 16x16 output tile; V_WMMA_F32_16X16X4_F32, K stepped by 4.
// Lane layouts per CDNA5 ISA 7.12.2:
//   A 16x4 : lane(0-15)=row M, VGPR{0,1}=K{0,1}; lanes 16-31 -> K{2,3}
//   B 4x16 : lane(0-15)=col N, VGPR{0,1}=K{0,1}; lanes 16-31 -> K{2,3}
//   D      : VGPR r -> rows {r, r+8}, lane&15 = col N
// All loads are arithmetically clamped in-bounds; masking is done with
// VALU selects on the loaded values so codegen emits straight
// global_load_b64 (no EXEC saveexec/branch sandwiches in the K-loop).
// =====================================================================
template<int K, int NOUT, bool RELU>
__global__ __launch_bounds__(256) void gemm_relu_wmma(
    const float* __restrict__ A, const float* __restrict__ W,
    const float* __restrict__ bias, float* __restrict__ C)
{
    static_assert((K & 1) == 0, "K must be even");
    constexpr int TILES_M = GG / 16;            // 256
    constexpr int TILES_N = (NOUT + 15) / 16;
    constexpr int TOTAL   = TILES_M * TILES_N;

    int wave = threadIdx.x >> 5;
    int lane = threadIdx.x & 31;
    int tile = blockIdx.x * 8 + wave;           // wave-uniform
    if (tile >= TOTAL) return;                  // uniform exit: EXEC all-1s for WMMA

    int tm = tile % TILES_M, tn = tile / TILES_M;
    int m0 = tm * 16, n0 = tn * 16;
    int row  = lane & 15;
    int half = lane >> 4;

    int m = m0 + row;                           // A row held by this lane
    int n = n0 + row;                           // B column / D column held by this lane
    bool nok = (n < NOUT);

    const float* Arow = A + (size_t)m * K;
    const float* Wrow = W + (size_t)(nok ? n : 0) * K;   // always a valid row

    v8f acc = {0.f, 0.f, 0.f, 0.f, 0.f, 0.f, 0.f, 0.f};

    for (int k0 = 0; k0 < K; k0 += 4) {
        int  ka  = k0 + half * 2;        // even; both elems valid iff ka < K
        bool kok = (ka < K);
        int  kc  = kok ? ka : 0;         // clamped, always in-bounds

        // unconditional in-bounds loads (-> global_load_b64 pairs)
        float a0 = Arow[kc], a1 = Arow[kc + 1];
        float b0 = Wrow[kc], b1 = Wrow[kc + 1];

        bool bok = nok && kok;
        v2f a, b;
        a[0] = kok ? a0 : 0.f;
        a[1] = kok ? a1 : 0.f;
        b[0] = bok ? b0 : 0.f;
        b[1] = bok ? b1 : 0.f;

        // 8 args: (neg_a, A, neg_b, B, c_mod, C, reuse_a, reuse_b)
        acc = __builtin_amdgcn_wmma_f32_16x16x4_f32(
                  false, a, false, b, (short)0, acc, false, false);
    }

    if (nok) {
        float bv = bias[n];
        #pragma unroll
        for (int r = 0; r < 8; ++r) {
            int   mm = m0 + half * 8 + r;
            float v  = acc[r] + bv;
            if (RELU) v = fmaxf(v, 0.f);
            C[(size_t)mm * NOUT + n] = v;
        }
    }
}

// =====================================================================
// Final 10->2 layer + numerically stable softmax, one thread per graph.
// =====================================================================
__global__ __launch_bounds__(256) void final_softmax_kernel(
    const float* __restrict__ H, const float* __restrict__ w,
    const float* __restrict__ b, float* __restrict__ out)
{
    int g = blockIdx.x * blockDim.x + threadIdx.x;
    if (g >= GG) return;
    float l0 = b[0], l1 = b[1];
    const float* h = H + (size_t)g * 10;
    #pragma unroll
    for (int k = 0; k < 10; ++k) {
        float hv = h[k];
        l0 += hv * w[k];
        l1 += hv * w[10 + k];
    }
    float mx = fmaxf(l0, l1);
    float e0 = expf(l0 - mx), e1 = expf(l1 - mx);
    float s  = e0 + e1;
    out[(size_t)g * 2 + 0] = e0 / s;
    out[(size_t)g * 2 + 1] = e1 / s;
}

// =====================================================================
extern "C" void kernel_launch(void* const* d_in, const int* in_sizes, int n_in,
                              void* d_out, int out_size, void* d_ws, size_t ws_size,
                              hipStream_t stream)
{
    (void)in_sizes; (void)n_in; (void)out_size; (void)ws_size;

    const float* x  = (const float*)d_in[0];
    const int*   ei = (const int*)  d_in[1];
    const float* ew = (const float*)d_in[2];
    const float *wr[5], *br[5], *wo[5], *lw[5], *lb[5];
    for (int i = 0; i < 5; ++i) {
        wr[i] = (const float*)d_in[3 + 3 * i];
        br[i] = (const float*)d_in[4 + 3 * i];
        wo[i] = (const float*)d_in[5 + 3 * i];
    }
    for (int i = 0; i < 5; ++i) {
        lw[i] = (const float*)d_in[18 + 2 * i];
        lb[i] = (const float*)d_in[19 + 2 * i];
    }

    float* ws = (float*)d_ws;
    float* H0 = ws;                              // [4096, 396]
    float* H1 = H0 + (size_t)GG * 396;           // [4096, 200]
    float* H2 = H1 + (size_t)GG * 200;           // [4096, 100]
    float* H3 = H2 + (size_t)GG * 100;           // [4096,  50]
    float* H4 = H3 + (size_t)GG * 50;            // [4096,  10]

    conv_fused_kernel<<<GG, 256, 0, stream>>>(
        x, ei, ew,
        wr[0], br[0], wo[0], wr[1], br[1], wo[1], wr[2], br[2], wo[2],
        wr[3], br[3], wo[3], wr[4], br[4], wo[4], H0);

    {   constexpr int T = (GG / 16) * ((200 + 15) / 16);
        gemm_relu_wmma<396, 200, true><<<(T + 7) / 8, 256, 0, stream>>>(H0, lw[0], lb[0], H1); }
    {   constexpr int T = (GG / 16) * ((100 + 15) / 16);
        gemm_relu_wmma<200, 100, true><<<(T + 7) / 8, 256, 0, stream>>>(H1, lw[1], lb[1], H2); }
    {   constexpr int T = (GG / 16) * ((50 + 15) / 16);
        gemm_relu_wmma<100, 50, true><<<(T + 7) / 8, 256, 0, stream>>>(H2, lw[2], lb[2], H3); }
    {   constexpr int T = (GG / 16) * ((10 + 15) / 16);
        gemm_relu_wmma<50, 10, true><<<(T + 7) / 8, 256, 0, stream>>>(H3, lw[3], lb[3], H4); }

    final_softmax_kernel<<<(GG + 255) / 256, 256, 0, stream>>>(
        H4, lw[4], lb[4], (float*)d_out);
}